// FusedSparseModules_14766097564310
// MI455X (gfx1250) — compile-verified
//
#include <hip/hip_runtime.h>

// ---------------------------------------------------------------------------
// EmbeddingBag(sum) over F*B bags, D=128, gathered from a concatenated table.
// HBM-bound (~0.24 FLOP/byte): wave-per-bag, 512B coalesced row gathers.
// Primary path: gfx1250 async global->LDS gathers (ASYNCcnt-tracked),
// triple-buffered pipeline with constant s_wait_asynccnt immediates so
// accumulation of group g overlaps the in-flight gathers of group g+1.
// Fallback: direct b128 gathers + global_prefetch_b8.
// ---------------------------------------------------------------------------

typedef float v4f __attribute__((ext_vector_type(4)));
typedef int   v4i __attribute__((ext_vector_type(4)));

#define NUM_F 26
#define NUM_B 4096
#define DIM   128
#define NUM_BAGS (NUM_F * NUM_B)          // 106496
#define WAVES_PER_BLOCK 8
#define THREADS (WAVES_PER_BLOCK * 32)    // 256
#define HALF 4                            // rows per async group
#define NBUF 3                            // triple buffer (resolves LDS WAR)

static_assert(NUM_B == 4096, "bag->(f,b) decomposition assumes B=4096");
static_assert(NUM_BAGS % WAVES_PER_BLOCK == 0, "exact grid");

#if defined(__has_builtin)
#if __has_builtin(__builtin_amdgcn_global_load_async_to_lds_b128) && \
    __has_builtin(__builtin_amdgcn_s_wait_asynccnt)
#define USE_ASYNC 1
#endif
#endif
#ifndef USE_ASYNC
#define USE_ASYNC 0
#endif

#if USE_ASYNC

__global__ __launch_bounds__(THREADS)
void embag_sum_async_pipe_kernel(const int* __restrict__ values,
                                 const int* __restrict__ offsets,
                                 const float* __restrict__ table,
                                 float* __restrict__ out)
{
    // 8 waves x 3 buffers x 4 rows x 512B = 48 KB per block.
    __shared__ v4f lds_rows[WAVES_PER_BLOCK][NBUF][HALF][32];

    const int waveInBlk = threadIdx.x >> 5;
    const int lane      = threadIdx.x & 31;
    // bag is wave-uniform: make it explicit so offsets go through SMEM.
    const int bag = __builtin_amdgcn_readfirstlane(
        (int)(blockIdx.x * WAVES_PER_BLOCK + waveInBlk));

    const int f = bag >> 12;          // bag / NUM_B
    const int b = bag & (NUM_B - 1);  // bag % NUM_B

    const int start = offsets[bag];
    const int end   = offsets[bag + 1];
    const int count = end - start;
    const int nGroups = (count + HALF - 1) >> 2;   // groups of up to HALF rows

    v4f acc = {0.f, 0.f, 0.f, 0.f};

    // 32 contiguous bag indices live in lane registers; refill per 32 rows.
    int curChunk = -1;
    int myIdx = 0;

    auto issueGroup = [&](int g) {
        const int rlo = g * HALF;
        const int rhi = min(count, rlo + HALF);
        v4f (&buf)[HALF][32] = lds_rows[waveInBlk][g % NBUF];
        for (int r = rlo; r < rhi; ++r) {
            const int c = r >> 5;
            if (c != curChunk) {                     // coalesced index refill
                curChunk = c;
                const int cb = c << 5;
                myIdx = (lane < count - cb) ? values[start + cb + lane] : 0;
            }
            const int row = __shfl(myIdx, r & 31, 32);
            const float* gp = table + (size_t)row * DIM + lane * 4;
            __builtin_amdgcn_global_load_async_to_lds_b128(
                (__attribute__((address_space(1))) v4i*)gp,
                (__attribute__((address_space(3))) v4i*)&buf[r - rlo][lane],
                /*offset=*/0, /*cpol=*/0);
        }
    };

    if (nGroups > 0) issueGroup(0);
    if (nGroups > 1) issueGroup(1);

    for (int gi = 0; gi < nGroups; ++gi) {
        // Async loads complete in issue order: once <= HALF remain, group gi
        // (issued >= 2 groups ago or 1 full group ago) has fully landed.
        if (gi + 2 < nGroups) __builtin_amdgcn_s_wait_asynccnt(HALF);
        else                  __builtin_amdgcn_s_wait_asynccnt(0);

        const int sz = min(count, gi * HALF + HALF) - gi * HALF;
        v4f (&buf)[HALF][32] = lds_rows[waveInBlk][gi % NBUF];
        for (int t = 0; t < sz; ++t)
            acc += buf[t][lane];                      // ds_load_b128

        if (gi + 2 < nGroups) issueGroup(gi + 2);     // refill buffer (gi+2)%3
    }

    // out[b, f, :]  (write-once -> non-temporal)
    v4f* o = (v4f*)out + ((size_t)b * NUM_F + f) * (DIM / 4) + lane;
    __builtin_nontemporal_store(acc, o);
}

#else  // !USE_ASYNC ------------------------------------------------------------

__global__ __launch_bounds__(THREADS)
void embag_sum_direct_kernel(const int* __restrict__ values,
                             const int* __restrict__ offsets,
                             const float* __restrict__ table,
                             float* __restrict__ out)
{
    const int lane = threadIdx.x & 31;
    const int bag  = __builtin_amdgcn_readfirstlane(
        (int)(blockIdx.x * WAVES_PER_BLOCK + (threadIdx.x >> 5)));

    const int f = bag >> 12;
    const int b = bag & (NUM_B - 1);

    const int start = offsets[bag];
    const int end   = offsets[bag + 1];

    v4f acc = {0.f, 0.f, 0.f, 0.f};

    for (int base = start; base < end; base += 32) {
        const int nIdx = min(32, end - base);
        int myIdx = 0;
        if (lane < nIdx) {
            myIdx = values[base + lane];  // coalesced index load
            __builtin_prefetch(table + (size_t)myIdx * DIM, 0, 0);
            __builtin_prefetch(table + (size_t)myIdx * DIM + 64, 0, 0);
        }
#pragma unroll 4
        for (int j = 0; j < nIdx; ++j) {
            const int row = __shfl(myIdx, j, 32);
            const v4f* p = (const v4f*)(table + (size_t)row * DIM) + lane;
            acc += *p;  // global_load_b128, 512B/wave
        }
    }

    v4f* o = (v4f*)out + ((size_t)b * NUM_F + f) * (DIM / 4) + lane;
    __builtin_nontemporal_store(acc, o);
}

#endif

extern "C" void kernel_launch(void* const* d_in, const int* in_sizes, int n_in,
                              void* d_out, int out_size, void* d_ws, size_t ws_size,
                              hipStream_t stream) {
    (void)in_sizes; (void)n_in; (void)d_ws; (void)ws_size; (void)out_size;

    const int*   values  = (const int*)d_in[0];   // N flat indices
    const int*   offsets = (const int*)d_in[1];   // F*B + 1 bag boundaries
    const float* table   = (const float*)d_in[2]; // [TOTAL_VOCAB, D]
    float*       out     = (float*)d_out;         // [B, F, D]

    const dim3 grid(NUM_BAGS / WAVES_PER_BLOCK);  // 13312 blocks, exact
#if USE_ASYNC
    embag_sum_async_pipe_kernel<<<grid, THREADS, 0, stream>>>(values, offsets, table, out);
#else
    embag_sum_direct_kernel<<<grid, THREADS, 0, stream>>>(values, offsets, table, out);
#endif
}